// QFTLayer_18726057410942
// MI455X (gfx1250) — compile-verified
//
#include <hip/hip_runtime.h>
#include <math.h>

// ---------------------------------------------------------------------------
// Batched 4096-pt QFT via four-step factorization 4096 = 64 x 64, computed
// with CDNA5 V_WMMA_F32_16X16X4_F32 (full f32 precision, D = A*B + C).
//
//   y = (1/64) * sum_k exp(+2*pi*i*j*k/4096) x[k]
//   X[k1][k2], k = k1*64 + k2
//   A = (F/8) @ X                  (F[j][k] = w[(j*k) & 63], w = exp(2*pi*i/64 * idx))
//   B[j1][k2] = tw(j1,k2) * A      (tw = exp(2*pi*i*j1*k2/4096))
//   C = B @ (F/8)                  (1/8 * 1/8 = 1/64 total normalization)
//   y[j1 + 64*j2] = C[j1][j2]      (store transposed)
//
// - DFT-64 factor = 64-entry complex LDS table (512 B), fragments built with
//   v_mul_u32_u24 + v_and + ds_load_b64 (no sincos in the hot loop).
// - X staged TRANSPOSED in LDS so stage-1 B-fragments are contiguous b64 loads
//   (no ds_load_2addr + v_mov repacking).
// - Pitch 68 (== 4 mod 64): both b64 fragment patterns are bank-conflict-free.
// ---------------------------------------------------------------------------

typedef float v2f __attribute__((ext_vector_type(2)));
typedef float v8f __attribute__((ext_vector_type(8)));

#define PITCH 68                       // LDS row pitch (floats); 68 % 64 == 4
#define TWO_PI_64 0.09817477042f       // 2*pi/64
#define TWO_PI_N  1.53398078789e-3f    // 2*pi/4096

#define WMMA_F32(A, B, C) \
  __builtin_amdgcn_wmma_f32_16x16x4_f32(false, (A), false, (B), (short)0, (C), false, false)

// 16x4 A-operand fragment: lane L<16 -> row rowBase+L, K pair {kb,kb+1};
// lane L>=16 -> K pair {kb+2,kb+3}. Contiguous 8B -> ds_load_b64.
// (Also serves as the B-operand loader from a TRANSPOSED plane, with
//  rowBase = column base and kb = K base.)
__device__ __forceinline__ v2f lds_afrag(const float* p, int lane, int rowBase, int kb) {
  const float* q = p + (rowBase + (lane & 15)) * PITCH + kb + 2 * (lane >> 4);
  v2f a; a.x = q[0]; a.y = q[1]; return a;
}

// DFT-64 factor fragments via the 64-entry complex root table (pre-scaled 1/8).
// A-operand layout: rows rowBase+(lane&15), K = kb+2*(lane>>4)+{0,1}.
__device__ __forceinline__ v2f dft_afrag(const float2* w, int lane, int rowBase, int kb, v2f* im) {
  const int r = rowBase + (lane & 15);
  const int k = kb + 2 * (lane >> 4);
  const float2 w0 = w[(r * k) & 63];
  const float2 w1 = w[(r * (k + 1)) & 63];
  v2f re; re.x = w0.x; re.y = w1.x;
  im->x = w0.y; im->y = w1.y;
  return re;
}

// B-operand layout: K = kb+2*(lane>>4)+{0,1}, col = colBase+(lane&15).
__device__ __forceinline__ v2f dft_bfrag(const float2* w, int lane, int kb, int colBase, v2f* im) {
  const int n = colBase + (lane & 15);
  const int k = kb + 2 * (lane >> 4);
  const float2 w0 = w[(k * n) & 63];
  const float2 w1 = w[((k + 1) * n) & 63];
  v2f re; re.x = w0.x; re.y = w1.x;
  im->x = w0.y; im->y = w1.y;
  return re;
}

__global__ __launch_bounds__(256) void qft4096_wmma_kernel(
    const float* __restrict__ gr, const float* __restrict__ gi,
    float* __restrict__ out, int nbatch) {
  __shared__ float  sR[64 * PITCH];
  __shared__ float  sI[64 * PITCH];
  __shared__ float2 sW[64];            // exp(2*pi*i*idx/64) / 8

  const int b    = blockIdx.x;
  const int tid  = threadIdx.x;
  const int lane = tid & 31;
  const int wave = tid >> 5;

  // ---- stage 0: root table + coalesced load of one state vector into LDS,
  //      stored TRANSPOSED: X_T[k2][k1] so stage-1 B-frags are contiguous b64.
  if (tid < 64) {
    float s, c;
    __sincosf((float)tid * TWO_PI_64, &s, &c);
    sW[tid] = make_float2(c * 0.125f, s * 0.125f);
  }
  const float* xr = gr + (size_t)b * 4096;
  const float* xi = gi + (size_t)b * 4096;
  for (int idx = tid; idx < 4096; idx += 256) {
    const int k1 = idx >> 6, k2 = idx & 63;
    sR[k2 * PITCH + k1] = __builtin_nontemporal_load(xr + idx);
    sI[k2 * PITCH + k1] = __builtin_nontemporal_load(xi + idx);
  }
  __syncthreads();

  // Tile assignment: wave w owns tiles (mT, nT0) and (mT, nT0+1), sharing the
  // A-operand (row) fragments between its two tiles.
  const int mBase = (wave >> 1) * 16;
  const int n0    = ((2 * wave) & 3) * 16;
  const int n1    = n0 + 16;

  const v8f vzero = {0.f, 0.f, 0.f, 0.f, 0.f, 0.f, 0.f, 0.f};

  // ---- stage 1: A = (F/8) @ X  (complex); X read from transposed plane
  v8f cr0 = vzero, ci0 = vzero, cr1 = vzero, ci1 = vzero;
  for (int kb = 0; kb < 64; kb += 4) {
    v2f fi_;
    v2f fr  = dft_afrag(sW, lane, mBase, kb, &fi_);
    v2f fin = {-fi_.x, -fi_.y};
    v2f x0r = lds_afrag(sR, lane, n0, kb);   // B-frag from X_T: contiguous b64
    v2f x0i = lds_afrag(sI, lane, n0, kb);
    v2f x1r = lds_afrag(sR, lane, n1, kb);
    v2f x1i = lds_afrag(sI, lane, n1, kb);
    cr0 = WMMA_F32(fr,  x0r, cr0);  cr0 = WMMA_F32(fin, x0i, cr0);
    ci0 = WMMA_F32(fr,  x0i, ci0);  ci0 = WMMA_F32(fi_, x0r, ci0);
    cr1 = WMMA_F32(fr,  x1r, cr1);  cr1 = WMMA_F32(fin, x1i, cr1);
    ci1 = WMMA_F32(fr,  x1i, ci1);  ci1 = WMMA_F32(fi_, x1r, ci1);
  }
  __syncthreads();  // all stage-1 reads of X complete before overwriting

  // ---- twiddle in registers, write B back to LDS (row-major: B[j1][k2])
  {
    const int h = lane >> 4, cc = lane & 15;
#pragma unroll
    for (int r = 0; r < 8; ++r) {
      const int m = mBase + r + 8 * h;
      {
        const int n = n0 + cc;
        float s, co;
        __sincosf((float)(m * n) * TWO_PI_N, &s, &co);
        const float ar = cr0[r], ai = ci0[r];
        sR[m * PITCH + n] = ar * co - ai * s;
        sI[m * PITCH + n] = ar * s + ai * co;
      }
      {
        const int n = n1 + cc;
        float s, co;
        __sincosf((float)(m * n) * TWO_PI_N, &s, &co);
        const float ar = cr1[r], ai = ci1[r];
        sR[m * PITCH + n] = ar * co - ai * s;
        sI[m * PITCH + n] = ar * s + ai * co;
      }
    }
  }
  __syncthreads();

  // ---- stage 2: C = B @ (F/8)  (complex); B read row-major (A-operand)
  v8f dr0 = vzero, di0 = vzero, dr1 = vzero, di1 = vzero;
  for (int kb = 0; kb < 64; kb += 4) {
    v2f br_ = lds_afrag(sR, lane, mBase, kb);
    v2f bi_ = lds_afrag(sI, lane, mBase, kb);
    v2f f0i, f1i;
    v2f f0r  = dft_bfrag(sW, lane, kb, n0, &f0i);
    v2f f1r  = dft_bfrag(sW, lane, kb, n1, &f1i);
    v2f f0in = {-f0i.x, -f0i.y};
    v2f f1in = {-f1i.x, -f1i.y};
    dr0 = WMMA_F32(br_, f0r, dr0);  dr0 = WMMA_F32(bi_, f0in, dr0);
    di0 = WMMA_F32(br_, f0i, di0);  di0 = WMMA_F32(bi_, f0r,  di0);
    dr1 = WMMA_F32(br_, f1r, dr1);  dr1 = WMMA_F32(bi_, f1in, dr1);
    di1 = WMMA_F32(br_, f1i, di1);  di1 = WMMA_F32(bi_, f1r,  di1);
  }
  __syncthreads();  // all stage-2 reads complete before overwriting

  // ---- write C transposed into LDS: Y[j2][j1] = C[j1][j2]
  {
    const int h = lane >> 4, cc = lane & 15;
#pragma unroll
    for (int r = 0; r < 8; ++r) {
      const int m = mBase + r + 8 * h;
      sR[(n0 + cc) * PITCH + m] = dr0[r];
      sI[(n0 + cc) * PITCH + m] = di0[r];
      sR[(n1 + cc) * PITCH + m] = dr1[r];
      sI[(n1 + cc) * PITCH + m] = di1[r];
    }
  }
  __syncthreads();

  // ---- coalesced non-temporal store: out[0,b,:,0] = real, out[1,b,:,0] = imag
  float* outR = out + (size_t)b * 4096;
  float* outI = out + (size_t)nbatch * 4096 + (size_t)b * 4096;
  for (int idx = tid; idx < 4096; idx += 256) {
    __builtin_nontemporal_store(sR[(idx >> 6) * PITCH + (idx & 63)], outR + idx);
    __builtin_nontemporal_store(sI[(idx >> 6) * PITCH + (idx & 63)], outI + idx);
  }
}

extern "C" void kernel_launch(void* const* d_in, const int* in_sizes, int n_in,
                              void* d_out, int out_size, void* d_ws, size_t ws_size,
                              hipStream_t stream) {
  const float* sr = (const float*)d_in[0];   // state_real  [B, 4096, 1] f32
  const float* si = (const float*)d_in[1];   // state_imag  [B, 4096, 1] f32
  float* out = (float*)d_out;                // [2, B, 4096, 1] f32

  const int nbatch = in_sizes[0] / 4096;     // 4096
  qft4096_wmma_kernel<<<dim3(nbatch), dim3(256), 0, stream>>>(sr, si, out, nbatch);
}